// Tagger_798863917282
// MI455X (gfx1250) — compile-verified
//
#include <hip/hip_runtime.h>

// ---------------------------------------------------------------------------
// Tagger scoring on MI455X (gfx1250).
//   out[w, l] = sum_{i: seg[i]==w} weights[fi[i], l] + trigram_w[clamp(sum_i fi[i])]
//
// Memory-bound: ~250MB HBM traffic -> ~11us floor @ 23.3 TB/s. The 90MB
// weights table is L2-resident (192MB L2), so the 720MB of gathered rows is
// L2 traffic. The segmented 45-wide reduction is a one-hot sparse x dense
// matmul on V_WMMA_F32_16X16X4_F32 (full f32, exact since A entries are
// 0.0/1.0), so the reduction rides the matrix pipe, not the VALU.
//
// Round-3 change vs round-2 (driven by the asm): offs[] loads for the tile
// bounds go through __builtin_amdgcn_readfirstlane, so `start`/`end` live in
// SGPRs. Loop control, the tail guard and the prefetch throttle become
// scalar branches (s_cmp/s_cbranch) instead of the v_cmp + exec-mask
// bookkeeping the compiler emitted for unprovably-uniform bounds — and EXEC
// is now provably all-ones at every WMMA, as the ISA requires.
// ---------------------------------------------------------------------------

typedef float v2f __attribute__((ext_vector_type(2)));
typedef float v8f __attribute__((ext_vector_type(8)));

#define N_LABELS 45
#define N_TRIGRAMS 4000000

__device__ __forceinline__ int imin(int a, int b) { return a < b ? a : b; }

// ---------------------------------------------------------------------------
// Kernel 1: CSR offsets from sorted segment ids. offs[w] = first index of
// word w; offs[nWords] = F. Handles empty words (runs of equal fills).
// ---------------------------------------------------------------------------
__global__ void build_offsets(const int* __restrict__ seg,
                              int* __restrict__ offs, int F, int nWords) {
  int i = blockIdx.x * blockDim.x + threadIdx.x;
  if (i >= F) return;
  int cur = seg[i];
  int prev = (i == 0) ? -1 : seg[i - 1];
  for (int w = prev + 1; w <= cur; ++w) offs[w] = i;
  if (i == F - 1) {
    for (int w = cur + 1; w <= nWords; ++w) offs[w] = F;
  }
}

// ---------------------------------------------------------------------------
// Kernel 2: per-word trigram scalar. tri_idx = sum of the word's feature
// indices, clamped like JAX's OOB gather clamp.
// ---------------------------------------------------------------------------
__global__ void trigram_kernel(const int* __restrict__ fi,
                               const int* __restrict__ offs,
                               const float* __restrict__ tw,
                               float* __restrict__ trival, int nWords) {
  int w = blockIdx.x * blockDim.x + threadIdx.x;
  if (w >= nWords) return;
  int s = offs[w], e = offs[w + 1];
  long long acc = 0;
  for (int i = s; i < e; ++i) acc += (long long)fi[i];
  if (acc < 0) acc = 0;
  if (acc > (long long)(N_TRIGRAMS - 1)) acc = N_TRIGRAMS - 1;
  trival[w] = tw[acc];
}

// ---------------------------------------------------------------------------
// Kernel 3: one wave32 per 16-word tile. For each chunk of 4 sorted features:
//   A[16x4] f32: A[m,k] = (seg[start+4c+k] == w0+m) ? 1 : 0
//   B[4x16]  f32: B[k,n] = weights[fi[start+4c+k]*45 + colbase + n]
//   D[16x16] += A x B   via v_wmma_f32_16x16x4_f32   (3 label tiles: 0,16,32)
// Layout (wave32): A 16x4 f32 in 2 VGPRs: lanes 0-15 = rows M=0..15 holding
// {K0,K1}; lanes 16-31 hold {K2,K3}. B mirrors the K split with N = lane&15.
// C/D: VGPR v = row v (lanes 0-15) / row v+8 (lanes 16-31), col = lane&15.
// Sorted segment ids => features past this tile's range fail the seg==word
// compare naturally; only the physical array end F needs a guard (tail only).
// ---------------------------------------------------------------------------
__global__ void tile_wmma(const int* __restrict__ fi,
                          const int* __restrict__ seg,
                          const float* __restrict__ W,
                          const float* __restrict__ trival,
                          const int* __restrict__ offs,
                          float* __restrict__ out, int nWords, int F) {
  const int lane = threadIdx.x & 31;
  const int wavesPerBlock = blockDim.x >> 5;
  const int tile = blockIdx.x * wavesPerBlock + (threadIdx.x >> 5);
  const int nTiles = (nWords + 15) >> 4;
  if (tile >= nTiles) return;  // wave-uniform exit: EXEC all-ones below

  const int w0 = tile << 4;
  // Force the tile bounds into SGPRs: scalar loop control + provably
  // all-ones EXEC at every WMMA.
  const int start = __builtin_amdgcn_readfirstlane(offs[w0]);
  const int end = __builtin_amdgcn_readfirstlane(offs[imin(w0 + 16, nWords)]);

  const int n = lane & 15;                 // B/D column within tile; A/D row m
  const int kbase = (lane < 16) ? 0 : 2;   // this lane's K pair
  const int wm = w0 + n;                   // word id this lane's A-row tracks
  const int col0 = n;                      // labels 0..15
  const int col1 = 16 + n;                 // labels 16..31
  const int col2 = imin(32 + n, N_LABELS - 1);  // labels 32..44 (clamped pad)

  v8f acc0 = {}, acc1 = {}, acc2 = {};

  int base = start;
  // ---- main loop: full chunks, no bounds checks, no clamps ----------------
  for (; base + 4 <= end; base += 4) {
    if (((base - start) & 31) == 0) {  // scalar branch, once per 128B line
      const int pf = imin(base + 256, F - 1);
      __builtin_prefetch(seg + pf, 0, 1);  // -> global_prefetch_b8
      __builtin_prefetch(fi + pf, 0, 1);
    }
    const int i0 = base + kbase;
    const int s0 = seg[i0];
    const int s1 = seg[i0 + 1];
    const int f0 = fi[i0] * N_LABELS;      // 32-bit offsets: saddr+scale form
    const int f1 = fi[i0 + 1] * N_LABELS;

    v2f a;
    a.x = (s0 == wm) ? 1.0f : 0.0f;
    a.y = (s1 == wm) ? 1.0f : 0.0f;

    v2f b0; b0.x = W[f0 + col0]; b0.y = W[f1 + col0];
    v2f b1; b1.x = W[f0 + col1]; b1.y = W[f1 + col1];
    v2f b2; b2.x = W[f0 + col2]; b2.y = W[f1 + col2];

    acc0 = __builtin_amdgcn_wmma_f32_16x16x4_f32(false, a, false, b0,
                                                 (short)0, acc0, false, false);
    acc1 = __builtin_amdgcn_wmma_f32_16x16x4_f32(false, a, false, b1,
                                                 (short)0, acc1, false, false);
    acc2 = __builtin_amdgcn_wmma_f32_16x16x4_f32(false, a, false, b2,
                                                 (short)0, acc2, false, false);
  }

  // ---- tail chunk: 1..3 features, branch-free predicate vs F --------------
  if (base < end) {  // scalar condition (SGPR bounds)
    const int i0 = base + kbase;
    const int i1 = i0 + 1;
    const int c0 = imin(i0, F - 1);  // clamp addresses; predicate kills value
    const int c1 = imin(i1, F - 1);
    const int s0 = seg[c0];
    const int s1 = seg[c1];
    const int f0 = fi[c0] * N_LABELS;
    const int f1 = fi[c1] * N_LABELS;

    v2f a;
    a.x = ((s0 == wm) & (i0 < F)) ? 1.0f : 0.0f;  // bitwise &: no branch
    a.y = ((s1 == wm) & (i1 < F)) ? 1.0f : 0.0f;

    v2f b0; b0.x = W[f0 + col0]; b0.y = W[f1 + col0];
    v2f b1; b1.x = W[f0 + col1]; b1.y = W[f1 + col1];
    v2f b2; b2.x = W[f0 + col2]; b2.y = W[f1 + col2];

    acc0 = __builtin_amdgcn_wmma_f32_16x16x4_f32(false, a, false, b0,
                                                 (short)0, acc0, false, false);
    acc1 = __builtin_amdgcn_wmma_f32_16x16x4_f32(false, a, false, b1,
                                                 (short)0, acc1, false, false);
    acc2 = __builtin_amdgcn_wmma_f32_16x16x4_f32(false, a, false, b2,
                                                 (short)0, acc2, false, false);
  }

  // ---- store D + per-word trigram scalar, 32-bit row offsets --------------
  const int mbase = (lane < 16) ? 0 : 8;
  const float* tv = trival + w0;                       // uniform SGPR base
  float* obase = out + (long long)w0 * N_LABELS;       // uniform SGPR base
#pragma unroll
  for (int v = 0; v < 8; ++v) {
    const int m = mbase + v;
    if (w0 + m >= nWords) continue;
    const float t = tv[m];
    const int ro = m * N_LABELS + n;
    obase[ro] = acc0[v] + t;
    obase[ro + 16] = acc1[v] + t;
    if (n < N_LABELS - 32) obase[ro + 32] = acc2[v] + t;
  }
}

// ---------------------------------------------------------------------------
// Launch. Inputs (setup_inputs order): feat_indices[i32], segment_ids[i32],
// weights[f32 500000x45], trigram_weights[f32 4M]. Output f32 [nWords x 45].
// Workspace: offs[nWords+1] ints, then trival[nWords] floats (~4MB).
// ---------------------------------------------------------------------------
extern "C" void kernel_launch(void* const* d_in, const int* in_sizes, int n_in,
                              void* d_out, int out_size, void* d_ws,
                              size_t ws_size, hipStream_t stream) {
  const int* fi = (const int*)d_in[0];
  const int* seg = (const int*)d_in[1];
  const float* W = (const float*)d_in[2];
  const float* tw = (const float*)d_in[3];
  float* out = (float*)d_out;

  const int F = in_sizes[0];
  const int nWords = out_size / N_LABELS;

  int* offs = (int*)d_ws;
  float* trival = (float*)((char*)d_ws + sizeof(int) * (size_t)(nWords + 1));

  build_offsets<<<(F + 255) / 256, 256, 0, stream>>>(seg, offs, F, nWords);
  trigram_kernel<<<(nWords + 255) / 256, 256, 0, stream>>>(fi, offs, tw,
                                                           trival, nWords);

  const int nTiles = (nWords + 15) / 16;
  const int wavesPerBlock = 8;  // 256 threads = 8 wave32
  const int blocks = (nTiles + wavesPerBlock - 1) / wavesPerBlock;
  tile_wmma<<<blocks, 32 * wavesPerBlock, 0, stream>>>(fi, seg, W, trival,
                                                       offs, out, nWords, F);
}